// RoPEMultiHeadSelfAttention_51092930953751
// MI455X (gfx1250) — compile-verified
//
#include <hip/hip_runtime.h>
#include <hip/hip_bf16.h>
#include <math.h>

// ---------------------------------------------------------------------------
// RoPE multi-head self-attention, MI455X (gfx1250), fp32 end-to-end.
// - All matmuls use V_WMMA_F32_16X16X4_F32 (full-precision WMMA): at ~137
//   GFLOP vs ~100 MB of HBM/L2 traffic the workload is bandwidth-bound on
//   MI455X, so fp32 WMMA is roofline-free and matches reference precision.
// - Global->LDS staging uses GLOBAL_LOAD_ASYNC_TO_LDS_B128 (ASYNCcnt) with
//   double buffering, so the next tile streams while WMMAs run.
// ---------------------------------------------------------------------------

typedef __attribute__((ext_vector_type(2))) float v2f;
typedef __attribute__((ext_vector_type(8))) float v8f;

#define D_MODEL 1024
#define SEQ_L   2048
#define N_HEAD  16
#define HEAD_D  64
#define N_ROWS  8192   // B * L

static __device__ __forceinline__ v8f wmma_f32(v2f a, v2f b, v8f c) {
  // D(16x16,f32) = A(16x4,f32) * B(4x16,f32) + C
  return __builtin_amdgcn_wmma_f32_16x16x4_f32(
      false, a, false, b, (short)0, c, false, false);
}

// Async 16B copy global -> LDS (tracked by ASYNCcnt). LDS destination byte
// address = low 32 bits of the generic pointer (ISA 10.2 aperture mapping).
static __device__ __forceinline__ void async_g2l_b128(const void* g, void* l) {
  unsigned loff = (unsigned)(size_t)l;
  asm volatile("global_load_async_to_lds_b128 %0, %1, off"
               :: "v"(loff), "v"(g) : "memory");
}

template <int N>
static __device__ __forceinline__ void wait_asynccnt() {
  asm volatile("s_wait_asynccnt %0" :: "i"(N) : "memory");
}

// ---------------------------------------------------------------------------
// GEMM: C[M,1024] = A[M,1024] * B[1024,1024], fp32 WMMA 16x16x4.
// Block = 256 threads (8 waves), block tile 128x128, wave tile 64x32.
// K consumed in 32-deep chunks staged in LDS, double-buffered with async
// global->LDS B128 copies pipelined via s_wait_asynccnt.
// LDS strides: A 36 floats (36m mod 64 is a permutation -> conflict-free
// A-fragment reads), B 132 floats; both multiples of 16 bytes for B128.
// ---------------------------------------------------------------------------
__global__ __launch_bounds__(256)
void gemm_f32_wmma(const float* __restrict__ A, const float* __restrict__ B,
                   float* __restrict__ C) {
  __shared__ float As[2][128][36];
  __shared__ float Bs[2][32][132];

  const int tid  = threadIdx.x;
  const int lane = tid & 31;
  const int w    = tid >> 5;
  const int nl   = lane & 15;
  const int hh   = lane >> 4;
  const int wr   = w >> 2;      // 0..1 : 64-row strip
  const int wc   = w & 3;       // 0..3 : 32-col strip
  const int m0   = blockIdx.y * 128;
  const int n0   = blockIdx.x * 128;

  auto issue = [&](int chunk, int buf) {
    const int kc = chunk * 32;
#pragma unroll
    for (int i = 0; i < 4; ++i) {
      const int s = tid + i * 256;
      { // A tile: 128 rows x 32 cols = 1024 float4 slots
        const int r = s >> 3, c4 = (s & 7) << 2;
        async_g2l_b128(A + (size_t)(m0 + r) * D_MODEL + kc + c4, &As[buf][r][c4]);
      }
      { // B tile: 32 rows x 128 cols = 1024 float4 slots
        const int r = s >> 5, c4 = (s & 31) << 2;
        async_g2l_b128(B + (size_t)(kc + r) * D_MODEL + n0 + c4, &Bs[buf][r][c4]);
      }
    }
  };

  v8f acc[4][2] = {};

  issue(0, 0);
  for (int c = 0; c < 32; ++c) {
    const int buf = c & 1;
    if (c + 1 < 32) { issue(c + 1, buf ^ 1); wait_asynccnt<8>(); }
    else            { wait_asynccnt<0>(); }
    __syncthreads();

#pragma unroll
    for (int kk = 0; kk < 8; ++kk) {
      v2f a[4];
#pragma unroll
      for (int rt = 0; rt < 4; ++rt)
        a[rt] = *(const v2f*)&As[buf][wr * 64 + rt * 16 + nl][kk * 4 + 2 * hh];
#pragma unroll
      for (int ctn = 0; ctn < 2; ++ctn) {
        v2f bf;
        bf.x = Bs[buf][kk * 4 + 2 * hh][wc * 32 + ctn * 16 + nl];
        bf.y = Bs[buf][kk * 4 + 2 * hh + 1][wc * 32 + ctn * 16 + nl];
#pragma unroll
        for (int rt = 0; rt < 4; ++rt)
          acc[rt][ctn] = wmma_f32(a[rt], bf, acc[rt][ctn]);
      }
    }
    __syncthreads();
  }

#pragma unroll
  for (int rt = 0; rt < 4; ++rt)
#pragma unroll
    for (int ctn = 0; ctn < 2; ++ctn)
#pragma unroll
      for (int g = 0; g < 8; ++g)
        C[(size_t)(m0 + wr * 64 + rt * 16 + g + 8 * hh) * D_MODEL +
          n0 + wc * 32 + ctn * 16 + nl] = acc[rt][ctn][g];
}

// ---------------------------------------------------------------------------
// Mask: mask[row] = (sum_d x[row,d] != 0) ? 1 : 0.   One wave per row.
// ---------------------------------------------------------------------------
__global__ __launch_bounds__(256)
void mask_kernel(const float* __restrict__ x, float* __restrict__ mask) {
  const int row  = blockIdx.x * 8 + (threadIdx.x >> 5);
  const int lane = threadIdx.x & 31;
  const float* xr = x + (size_t)row * D_MODEL;
  float s = 0.f;
#pragma unroll 4
  for (int j = lane; j < D_MODEL; j += 32) s += xr[j];
#pragma unroll
  for (int m = 1; m < 32; m <<= 1) s += __shfl_xor(s, m, 32);
  if (lane == 0) mask[row] = (s != 0.0f) ? 1.0f : 0.0f;
}

// ---------------------------------------------------------------------------
// RoPE, in place on Q and K. One thread = one (even,odd) pair.
// ---------------------------------------------------------------------------
__global__ __launch_bounds__(256)
void rope_kernel(float* __restrict__ q, float* __restrict__ k) {
  const int p    = blockIdx.x * 256 + threadIdx.x;  // 0 .. 8192*512-1
  const int row  = p >> 9;
  const int j    = p & 511;
  const int head = j >> 5;
  const int i    = j & 31;
  const int l    = row & (SEQ_L - 1);

  const float freq = __expf(-(float)(2 * i) * (9.210340371976184f / 64.0f));
  const float ang  = (float)l * freq;
  const float c = cosf(ang), s = sinf(ang);

  const size_t base = (size_t)row * D_MODEL + head * HEAD_D + 2 * i;
  float q1 = q[base], q2 = q[base + 1];
  q[base]     = q1 * c - q2 * s;
  q[base + 1] = q1 * s + q2 * c;
  float k1 = k[base], k2 = k[base + 1];
  k[base]     = k1 * c - k2 * s;
  k[base + 1] = k1 * s + k2 * c;
}

// ---------------------------------------------------------------------------
// Flash attention, fp32 WMMA. Grid = (L/64, H, B), block = 128 threads
// (4 waves). Wave w owns Q rows [qb*64 + w*16, +16). K/V streamed in
// double-buffered 64-key LDS tiles via async global->LDS; online softmax in
// the C-fragment layout; P staged through per-wave LDS for O += P*V.
// ---------------------------------------------------------------------------
__global__ __launch_bounds__(128)
void flash_attn_kernel(const float* __restrict__ Q, const float* __restrict__ K,
                       const float* __restrict__ V, const float* __restrict__ maskb,
                       float* __restrict__ Oout) {
  __shared__ float Ks[2][64][68];
  __shared__ float Vs[2][64][68];
  __shared__ float Ps[4][16][68];

  const int b  = blockIdx.z;
  const int h  = blockIdx.y;
  const int qb = blockIdx.x;
  const int tid  = threadIdx.x;
  const int w    = tid >> 5;
  const int lane = tid & 31;
  const int nl   = lane & 15;
  const int hh   = lane >> 4;
  const int hoff = h * HEAD_D;
  const size_t rowQ = (size_t)b * SEQ_L + qb * 64 + w * 16;

  auto issue = [&](int kb, int buf) {
    const size_t rowK = (size_t)b * SEQ_L + kb * 64;
#pragma unroll
    for (int i = 0; i < 8; ++i) {
      const int s = tid + i * 128;             // 1024 float4 slots per tile
      const int r = s >> 4, c4 = (s & 15) << 2;
      async_g2l_b128(&K[(rowK + r) * D_MODEL + hoff + c4], &Ks[buf][r][c4]);
      async_g2l_b128(&V[(rowK + r) * D_MODEL + hoff + c4], &Vs[buf][r][c4]);
    }
  };

  // Persistent Q A-fragments: 16 k-steps of 16x4.
  v2f qa[16];
  const float* qrow = Q + (rowQ + nl) * D_MODEL + hoff + 2 * hh;
#pragma unroll
  for (int ks = 0; ks < 16; ++ks) qa[ks] = *(const v2f*)(qrow + ks * 4);

  v8f o[4] = {};
  float mrow[8], lrow[8];
#pragma unroll
  for (int g = 0; g < 8; ++g) { mrow[g] = -3.0e38f; lrow[g] = 0.0f; }

  issue(0, 0);
  for (int kb = 0; kb < SEQ_L / 64; ++kb) {
    const int buf = kb & 1;
    if (kb + 1 < SEQ_L / 64) { issue(kb + 1, buf ^ 1); wait_asynccnt<16>(); }
    else                     { wait_asynccnt<0>(); }
    __syncthreads();

    // S = Q * K^T  (4 col tiles of 16, k-dim = 64)
    v8f sf[4];
#pragma unroll
    for (int ct = 0; ct < 4; ++ct) {
      v8f accs = {};
#pragma unroll
      for (int ks = 0; ks < 16; ++ks) {
        v2f bf = *(const v2f*)&Ks[buf][ct * 16 + nl][ks * 4 + 2 * hh];
        accs = wmma_f32(qa[ks], bf, accs);
      }
      sf[ct] = accs;
    }

    // scale + additive mask (per key column)
    float mk[4];
#pragma unroll
    for (int ct = 0; ct < 4; ++ct)
      mk[ct] = maskb[b * SEQ_L + kb * 64 + ct * 16 + nl];
#pragma unroll
    for (int ct = 0; ct < 4; ++ct)
#pragma unroll
      for (int g = 0; g < 8; ++g)
        sf[ct][g] = sf[ct][g] * 0.125f + mk[ct];   // 1/sqrt(64)

    // Online softmax per row g (rows live in 16-lane half-waves)
#pragma unroll
    for (int g = 0; g < 8; ++g) {
      float rmax = fmaxf(fmaxf(sf[0][g], sf[1][g]), fmaxf(sf[2][g], sf[3][g]));
#pragma unroll
      for (int mm = 1; mm < 16; mm <<= 1) rmax = fmaxf(rmax, __shfl_xor(rmax, mm, 32));
      const float mnew  = fmaxf(mrow[g], rmax);
      const float alpha = __expf(mrow[g] - mnew);
      mrow[g] = mnew;
      float rsum = 0.f;
#pragma unroll
      for (int ct = 0; ct < 4; ++ct) {
        const float p = __expf(sf[ct][g] - mnew);
        sf[ct][g] = p;
        rsum += p;
      }
#pragma unroll
      for (int mm = 1; mm < 16; mm <<= 1) rsum += __shfl_xor(rsum, mm, 32);
      lrow[g] = lrow[g] * alpha + rsum;
#pragma unroll
      for (int ct = 0; ct < 4; ++ct) o[ct][g] *= alpha;
#pragma unroll
      for (int ct = 0; ct < 4; ++ct) Ps[w][g + 8 * hh][ct * 16 + nl] = sf[ct][g];
    }

    // O += P * V  (P re-read from LDS in A-fragment layout; same-wave LDS
    // ops are in-order, so no barrier needed for the per-wave Ps tile)
#pragma unroll
    for (int ks = 0; ks < 16; ++ks) {
      v2f pa = *(const v2f*)&Ps[w][nl][ks * 4 + 2 * hh];
#pragma unroll
      for (int ct = 0; ct < 4; ++ct) {
        v2f bv;
        bv.x = Vs[buf][ks * 4 + 2 * hh][ct * 16 + nl];
        bv.y = Vs[buf][ks * 4 + 2 * hh + 1][ct * 16 + nl];
        o[ct] = wmma_f32(pa, bv, o[ct]);
      }
    }
    __syncthreads();
  }

#pragma unroll
  for (int ct = 0; ct < 4; ++ct)
#pragma unroll
    for (int g = 0; g < 8; ++g)
      Oout[(rowQ + g + 8 * hh) * D_MODEL + hoff + ct * 16 + nl] = o[ct][g] / lrow[g];
}

// ---------------------------------------------------------------------------
// Launch pipeline (all on `stream`; no allocs, graph-capture safe).
// ---------------------------------------------------------------------------
extern "C" void kernel_launch(void* const* d_in, const int* in_sizes, int n_in,
                              void* d_out, int out_size, void* d_ws, size_t ws_size,
                              hipStream_t stream) {
  const float* x  = (const float*)d_in[0];
  const float* wq = (const float*)d_in[1];
  const float* wk = (const float*)d_in[2];
  const float* wv = (const float*)d_in[3];
  const float* wo = (const float*)d_in[4];
  float* out = (float*)d_out;

  char* ws = (char*)d_ws;
  const size_t MAT = (size_t)N_ROWS * D_MODEL * sizeof(float);  // 32 MB
  float* q    = (float*)(ws + 0 * MAT);
  float* k    = (float*)(ws + 1 * MAT);
  float* v    = (float*)(ws + 2 * MAT);
  float* attn = (float*)(ws + 3 * MAT);
  float* mask = (float*)(ws + 4 * MAT);

  dim3 gemm_grid(D_MODEL / 128, N_ROWS / 128);   // (8, 64)
  gemm_f32_wmma<<<gemm_grid, 256, 0, stream>>>(x, wq, q);
  gemm_f32_wmma<<<gemm_grid, 256, 0, stream>>>(x, wk, k);
  gemm_f32_wmma<<<gemm_grid, 256, 0, stream>>>(x, wv, v);

  mask_kernel<<<N_ROWS / 8, 256, 0, stream>>>(x, mask);

  rope_kernel<<<(N_ROWS * (D_MODEL / 2)) / 256, 256, 0, stream>>>(q, k);

  dim3 attn_grid(SEQ_L / 64, N_HEAD, 4);         // (32, 16, 4)
  flash_attn_kernel<<<attn_grid, 128, 0, stream>>>(q, k, v, mask, attn);

  gemm_f32_wmma<<<gemm_grid, 256, 0, stream>>>(attn, wo, out);
}